// EmotionModel_40561671144053
// MI455X (gfx1250) — compile-verified
//
#include <hip/hip_runtime.h>

typedef __attribute__((ext_vector_type(2))) float v2f;
typedef __attribute__((ext_vector_type(8))) float v8f;

#define D 7
#define TPB 256

// ---------------------------------------------------------------------------
// Setup kernel: one wave32. Computes
//   ws[0..48]  = M = W_Q^T @ W_K   (7x7, row-major)  via V_WMMA_F32_16X16X4_F32
//   ws[49..55] = colsum(W_D)
//   ws[56]     = sum(b_D)
// A-matrix (16x4 f32): lane m%16 = row, (VGPR v, lane/16) -> K = v + 2*(lane/16)
// B-matrix (4x16 f32): (VGPR v, lane/16) -> K = v + 2*(lane/16), lane%16 = col
// D-matrix (16x16 f32): VGPR g -> row (g + 8*(lane/16)), lane%16 = col
// ---------------------------------------------------------------------------
__global__ void emotion_setup_kernel(const float* __restrict__ WQ,
                                     const float* __restrict__ WK,
                                     const float* __restrict__ WD,
                                     const float* __restrict__ bD,
                                     float* __restrict__ ws) {
    const int lane = threadIdx.x;      // 0..31, all active (EXEC all-1s for WMMA)
    const int m    = lane & 15;        // row (A) / col (B)
    const int half = lane >> 4;        // 0 or 1

    v8f c = {};
#pragma unroll
    for (int iter = 0; iter < 2; ++iter) {
        v2f a, b;
#pragma unroll
        for (int v = 0; v < 2; ++v) {
            const int j = v + 2 * half + 4 * iter;   // K index (contraction dim)
            // A[m][j] = (W_Q^T)[m][j] = W_Q[j][m]
            a[v] = (m < D && j < D) ? WQ[j * D + m] : 0.0f;
            // B[j][n] = W_K[j][n], n = m
            b[v] = (m < D && j < D) ? WK[j * D + m] : 0.0f;
        }
        // D = A*B + C : accumulate K=0..3 then K=4..7 (row/col 7 zero-padded)
        c = __builtin_amdgcn_wmma_f32_16x16x4_f32(
            /*neg_a=*/false, a, /*neg_b=*/false, b,
            /*c_mod=*/(short)0, c, /*reuse_a=*/false, /*reuse_b=*/false);
    }

    // Write the 7x7 result: lanes 0..6 hold cols 0..6; VGPR g holds row g.
    if (half == 0 && m < D) {
#pragma unroll
        for (int g = 0; g < D; ++g) ws[g * D + m] = c[g];
    }

    // colsum(W_D) and sum(b_D)
    if (lane < D) {
        float acc = 0.0f;
#pragma unroll
        for (int j = 0; j < D; ++j) acc += WD[j * D + lane];
        ws[49 + lane] = acc;
    }
    if (lane == 0) {
        float acc = 0.0f;
#pragma unroll
        for (int j = 0; j < D; ++j) acc += bD[j];
        ws[56] = acc;
    }
}

// ---------------------------------------------------------------------------
// Streaming kernel: one thread = one batch row (7 floats in, 7 out).
// Global traffic is fully coalesced b32 loads/stores staged through LDS;
// per-thread LDS reads stride 7 dwords -> bank-conflict-free (gcd(7,64)=1).
// ---------------------------------------------------------------------------
__global__ void emotion_main_kernel(const float* __restrict__ rep,
                                    const float* __restrict__ emo,
                                    const float* __restrict__ ws,
                                    float* __restrict__ out,
                                    int Btotal) {
    __shared__ float ldsE[TPB * D];
    __shared__ float ldsR[TPB * D];

    const int t         = threadIdx.x;
    const int rowStart  = blockIdx.x * TPB;
    const int rows      = (Btotal - rowStart < TPB) ? (Btotal - rowStart) : TPB;
    const int base      = rowStart * D;
    const int total     = rows * D;

    // Stage inputs: coalesced dword loads
    for (int idx = t; idx < total; idx += TPB) {
        ldsE[idx] = emo[base + idx];
        ldsR[idx] = rep[base + idx];
    }
    __syncthreads();

    float e[D], r[D];
    if (t < rows) {
#pragma unroll
        for (int k = 0; k < D; ++k) {
            e[k] = ldsE[t * D + k];
            r[k] = ldsR[t * D + k];
        }
    }
    __syncthreads();   // everyone done reading ldsE before reuse as output stage

    if (t < rows) {
        // Uniform constants -> scalar loads
        float t7[D];
#pragma unroll
        for (int i = 0; i < D; ++i) t7[i] = 0.0f;
#pragma unroll
        for (int k = 0; k < D; ++k) {
#pragma unroll
            for (int i = 0; i < D; ++i)
                t7[i] = __builtin_fmaf(e[k], ws[k * D + i], t7[i]);
        }

        // raw = r * t ; softmax over the 7 entries
        float raw[D];
        float mx = -__builtin_inff();
#pragma unroll
        for (int i = 0; i < D; ++i) {
            raw[i] = r[i] * t7[i];
            mx = fmaxf(mx, raw[i]);
        }
        float s[D], sum = 0.0f;
#pragma unroll
        for (int i = 0; i < D; ++i) {
            s[i] = __expf(raw[i] - mx);
            sum += s[i];
        }
        const float inv  = 1.0f / sum;
        const float bsum = ws[56];
#pragma unroll
        for (int i = 0; i < D; ++i) {
            const float si = s[i] * inv;
            float d = si * si * si * ws[49 + i] + bsum;
            d = fminf(1.0f, fmaxf(-1.0f, d));
            ldsE[t * D + i] = r[i] + d;   // stage result
        }
    }
    __syncthreads();

    // Coalesced dword stores
    for (int idx = t; idx < total; idx += TPB) {
        out[base + idx] = ldsE[idx];
    }
}

// ---------------------------------------------------------------------------
extern "C" void kernel_launch(void* const* d_in, const int* in_sizes, int n_in,
                              void* d_out, int out_size, void* d_ws, size_t ws_size,
                              hipStream_t stream) {
    const float* rep = (const float*)d_in[0];
    const float* emo = (const float*)d_in[1];
    const float* WQ  = (const float*)d_in[2];
    const float* WK  = (const float*)d_in[3];
    const float* WD  = (const float*)d_in[4];
    const float* bD  = (const float*)d_in[5];
    float*       out = (float*)d_out;
    float*       ws  = (float*)d_ws;   // needs 57 floats

    const int Btotal = in_sizes[0] / D;

    emotion_setup_kernel<<<1, 32, 0, stream>>>(WQ, WK, WD, bD, ws);

    const int grid = (Btotal + TPB - 1) / TPB;
    emotion_main_kernel<<<grid, TPB, 0, stream>>>(rep, emo, ws, out, Btotal);
}